// F1InstanceLoss2dIntersection_65317862638317
// MI455X (gfx1250) — compile-verified
//
#include <hip/hip_runtime.h>
#include <hip/hip_bf16.h>

#define HH 512
#define WW 512
#define NIMG 64          // B*C = 16*4
#define NT 3             // tensors: 0=inter, 1=input, 2=target
#define STRIPES 16
#define RPS (HH / STRIPES)   // 32 rows per stripe
#define NWAVES 16            // 512 threads / wave32

typedef __attribute__((ext_vector_type(2))) float v2f;
typedef __attribute__((ext_vector_type(8))) float v8f;

// ---------------------------------------------------------------------------
// Stage 1: streaming pass over both inputs. Produces, per image and tensor:
//   gx[col] = sum_h |x[h+1,col] - x[h-1,col]|   (atomic-accumulated over stripes)
//   gy[row] = sum_w |x[row,w+1] - x[row,w-1]|   (plain store, rows partitioned)
// ---------------------------------------------------------------------------
__global__ __launch_bounds__(512)
void f1loss_stage1(const float* __restrict__ inp, const float* __restrict__ tgt,
                   float* __restrict__ gx, float* __restrict__ gy) {
  const int img = blockIdx.y;
  const int r0  = blockIdx.x * RPS;
  const int t   = threadIdx.x;      // column owned by this thread
  const int lane = t & 31;
  const int wv   = t >> 5;

  const float* A  = inp + (size_t)img * HH * WW;
  const float* Bp = tgt + (size_t)img * HH * WW;

  __shared__ float gyw[NWAVES][NT][RPS];

  // vertical (gx) accumulators per column, per tensor (p = product/inter)
  float gpa = 0.f, gia = 0.f, gta = 0.f;
  // row history: *_1 = row r-1, *_2 = row r-2
  float a1 = 0.f, b1 = 0.f, p1 = 0.f;
  float a2 = 0.f, b2 = 0.f, p2 = 0.f;

  for (int r = r0 - 1; r <= r0 + RPS; ++r) {
    float av = 0.f, bv = 0.f;
    if ((unsigned)r < (unsigned)HH) {
      av = A[(size_t)r * WW + t];
      bv = Bp[(size_t)r * WW + t];
    }
    float pv = av * bv;

    // horizontal terms for rows belonging to this stripe
    if (r >= r0 && r < r0 + RPS) {
      float am = 0.f, ap = 0.f, bm = 0.f, bpv = 0.f;
      if (t > 0)      { am  = A[(size_t)r * WW + t - 1]; bm  = Bp[(size_t)r * WW + t - 1]; }
      if (t < WW - 1) { ap  = A[(size_t)r * WW + t + 1]; bpv = Bp[(size_t)r * WW + t + 1]; }
      float hp = fabsf(ap * bpv - am * bm);
      float hi = fabsf(ap - am);
      float ht = fabsf(bpv - bm);
      #pragma unroll
      for (int off = 16; off > 0; off >>= 1) {
        hp += __shfl_down(hp, off);
        hi += __shfl_down(hi, off);
        ht += __shfl_down(ht, off);
      }
      if (lane == 0) {
        const int rl = r - r0;
        gyw[wv][0][rl] = hp;
        gyw[wv][1][rl] = hi;
        gyw[wv][2][rl] = ht;
      }
    }

    // vertical term centered at row c = r-1: |x[c+1] - x[c-1]|
    const int c = r - 1;
    if (c >= r0 && c < r0 + RPS) {
      gpa += fabsf(pv - p2);
      gia += fabsf(av - a2);
      gta += fabsf(bv - b2);
    }
    a2 = a1; b2 = b1; p2 = p1;
    a1 = av; b1 = bv; p1 = pv;
  }

  // gx: sum partial column profiles across stripes
  atomicAdd(&gx[((size_t)img * NT + 0) * WW + t], gpa);
  atomicAdd(&gx[((size_t)img * NT + 1) * WW + t], gia);
  atomicAdd(&gx[((size_t)img * NT + 2) * WW + t], gta);

  __syncthreads();
  // gy: fold the 16 per-wave partials; each row is owned by exactly one block
  if (t < NT * RPS) {
    const int k = t / RPS, rl = t % RPS;
    float s = 0.f;
    #pragma unroll
    for (int w = 0; w < NWAVES; ++w) s += gyw[w][k][rl];
    gy[((size_t)img * NT + k) * HH + (r0 + rl)] = s;
  }
}

// ---------------------------------------------------------------------------
// Stage 2: E = (t[0]+...+t[511]) / 16 where
//   t[0]=g[0], t[1]=g[1], t[511]=g[509], t[l]=|g[l]-g[l-2]| otherwise.
// 16 independent arrays per 32-thread block; summed with chained
// v_wmma_f32_16x16x4_f32 against an all-ones B (K marches 4/step, C chains).
// A layout (16x4 f32): lane L holds A[M = L%16, K = (L/16)*2 + vgpr].
// D layout: vgpr r, lanes 0-15 -> (M=r, N=lane); lanes 16-31 -> (M=8+r).
// ---------------------------------------------------------------------------
__global__ __launch_bounds__(32)
void f1loss_stage2(const float* __restrict__ gx, const float* __restrict__ gy,
                   float* __restrict__ E) {
  __shared__ float terms[16][WW];
  const int lane = threadIdx.x;
  const int base = blockIdx.x * 16;   // global array index: a = ((img*3)+k)*2 + dir

  for (int la = 0; la < 16; ++la) {
    const int a = base + la;
    const float* g = (a & 1) ? (gy + (size_t)(a >> 1) * HH)
                             : (gx + (size_t)(a >> 1) * WW);
    for (int i = lane; i < WW; i += 32) {
      float tv;
      if (i == 0)            tv = g[0];
      else if (i == 1)       tv = g[1];
      else if (i == WW - 1)  tv = g[WW - 3];
      else                   tv = fabsf(g[i] - g[i - 2]);
      terms[la][i] = tv;
    }
  }
  __syncthreads();

  const int m    = lane & 15;
  const int koff = (lane >> 4) * 2;
  v8f acc = {0.f, 0.f, 0.f, 0.f, 0.f, 0.f, 0.f, 0.f};
  v2f ones; ones.x = 1.0f; ones.y = 1.0f;

  for (int c4 = 0; c4 < WW / 4; ++c4) {
    v2f av;
    av.x = terms[m][4 * c4 + koff];
    av.y = terms[m][4 * c4 + koff + 1];
    // D = A * ones + C  ->  D[m,n] += terms[m][4c..4c+3]
    acc = __builtin_amdgcn_wmma_f32_16x16x4_f32(
        false, av, false, ones, (short)0, acc, false, false);
  }

  const float inv16 = 1.0f / 16.0f;
  if (lane == 0) {            // (M = 0..7, N = 0)
    #pragma unroll
    for (int r = 0; r < 8; ++r) E[base + r] = acc[r] * inv16;
  } else if (lane == 16) {    // (M = 8..15, N = 0)
    #pragma unroll
    for (int r = 0; r < 8; ++r) E[base + 8 + r] = acc[r] * inv16;
  }
}

// ---------------------------------------------------------------------------
// Stage 3: F1 arithmetic per (b,c) and batch mean.
// ---------------------------------------------------------------------------
__global__ __launch_bounds__(64)
void f1loss_stage3(const float* __restrict__ E, float* __restrict__ out) {
  __shared__ float red[NIMG];
  const int i = threadIdx.x;            // image index = b*C + c
  const float* e = E + (size_t)i * NT * 2;
  const float ie = e[0] + e[1];         // inter
  const float xe = e[2] + e[3];         // input
  const float te = e[4] + e[5];         // target
  const float mism = fmaxf(ie - xe, 0.f);
  const float tp = fabsf(ie - mism);
  const float fn = fmaxf(te - ie, 0.f) + mism;
  const float fp = fmaxf(xe - ie, 0.f);
  red[i] = 1.f - (2.f * tp + 1.f) / (2.f * tp + fp + fn + 1.f);
  __syncthreads();
  if (i == 0) {
    float s = 0.f;
    for (int j = 0; j < NIMG; ++j) s += red[j];
    out[0] = s * (1.0f / 16.0f);        // mean over B=16 of per-b channel sums
  }
}

extern "C" void kernel_launch(void* const* d_in, const int* in_sizes, int n_in,
                              void* d_out, int out_size, void* d_ws, size_t ws_size,
                              hipStream_t stream) {
  const float* inp = (const float*)d_in[0];
  const float* tgt = (const float*)d_in[1];

  float* ws = (float*)d_ws;
  float* gx = ws;                                   // NIMG*NT*WW floats
  float* gy = gx + (size_t)NIMG * NT * WW;          // NIMG*NT*HH floats
  float* E  = gy + (size_t)NIMG * NT * HH;          // NIMG*NT*2 floats

  // gx is accumulated with atomics -> must start at zero each call
  hipMemsetAsync(gx, 0, (size_t)NIMG * NT * WW * sizeof(float), stream);

  dim3 g1(STRIPES, NIMG);
  f1loss_stage1<<<g1, 512, 0, stream>>>(inp, tgt, gx, gy);
  f1loss_stage2<<<(NIMG * NT * 2) / 16, 32, 0, stream>>>(gx, gy, E);
  f1loss_stage3<<<1, 64, 0, stream>>>(E, (float*)d_out);
}